// DeformationGraph_geod_71287867179597
// MI455X (gfx1250) — compile-verified
//
#include <hip/hip_runtime.h>

typedef float v2f __attribute__((ext_vector_type(2)));
typedef float v8f __attribute__((ext_vector_type(8)));

#define N_VERTS_C 1000000
#define N_NODES_C 500000
#define K_INFL    3
#define NEIGH_C   9

// ---------------------------------------------------------------------------
// Wave-level reduction of TWO independent f32 partial sums (one pair per lane)
// using a single V_WMMA_F32_16X16X4_F32.
//
// A (16x4 f32) layout (ISA 7.12.2): VGPR0 = K0 (lanes 0-15) / K2 (lanes 16-31),
//                                   VGPR1 = K1 (lanes 0-15) / K3 (lanes 16-31).
//   a.x = arap partial  ->  A[m][0] = arap[m],  A[m][2] = arap[m+16]
//   a.y = sr   partial  ->  A[m][1] = sr[m],    A[m][3] = sr[m+16]
// B (4x16 f32) selector: B[0][0]=B[2][0]=1, B[1][1]=B[3][1]=1, else 0
//   => D[m][0] = arap[m]+arap[m+16], D[m][1] = sr[m]+sr[m+16]
// C/D layout: lane n (n<16) holds column n, M=0..7 in v[0..7]; lane n+16 holds
// M=8..15. So lanes {0,16} carry the 16 arap row-sums, lanes {1,17} the sr ones.
// Requires EXEC == all ones: call only from fully-converged code.
// ---------------------------------------------------------------------------
__device__ __forceinline__ void wave_reduce2_wmma(float arap, float sr,
                                                  double* arap_acc, double* sr_acc) {
    const unsigned lane = threadIdx.x & 31u;

    v2f a;
    a[0] = arap;
    a[1] = sr;

    v2f b;
    b[0] = (lane == 0u || lane == 16u) ? 1.0f : 0.0f;   // rows K=0, K=2 -> column 0
    b[1] = (lane == 1u || lane == 17u) ? 1.0f : 0.0f;   // rows K=1, K=3 -> column 1

    v8f c = {0.f, 0.f, 0.f, 0.f, 0.f, 0.f, 0.f, 0.f};
    v8f d = __builtin_amdgcn_wmma_f32_16x16x4_f32(
        /*neg_a=*/false, a, /*neg_b=*/false, b,
        /*c_mod=*/(short)0, c, /*reuse_a=*/false, /*reuse_b=*/false);

    float s = d[0] + d[1] + d[2] + d[3] + d[4] + d[5] + d[6] + d[7];
    s += __shfl_xor(s, 16, 32);   // combine lane pairs (0,16) and (1,17)

    if (lane == 0u)      atomicAdd(arap_acc, (double)s);
    else if (lane == 1u) atomicAdd(sr_acc,   (double)s);
}

// ---------------------------------------------------------------------------
// Zero the two f64 loss accumulators in workspace.
// ---------------------------------------------------------------------------
__global__ void init_kernel(double* acc) {
    if (threadIdx.x == 0 && blockIdx.x == 0) {
        acc[0] = 0.0;   // arap sum
        acc[1] = 0.0;   // sr sum
    }
}

// ---------------------------------------------------------------------------
// Warp kernel: one thread per vertex.
// out[n] = sum_k w[n,k] * ( R[i] @ (v[n] - p[i]) + p[i] + T[i] ),  i = infl[n,k]
// ---------------------------------------------------------------------------
__global__ void __launch_bounds__(256) warp_kernel(
    const float* __restrict__ vertices,
    const float* __restrict__ R,
    const float* __restrict__ T,
    const float* __restrict__ weights,
    const int*   __restrict__ nodes_idx,
    const int*   __restrict__ infl_idx,
    float*       __restrict__ out)
{
    const int n = blockIdx.x * blockDim.x + threadIdx.x;
    if (n >= N_VERTS_C) return;

    const float vx = vertices[3 * n + 0];
    const float vy = vertices[3 * n + 1];
    const float vz = vertices[3 * n + 2];

    float ox = 0.f, oy = 0.f, oz = 0.f;

#pragma unroll
    for (int k = 0; k < K_INFL; ++k) {
        const int ni = infl_idx[3 * n + k];
        const int vi = nodes_idx[ni];

        const float px = vertices[3 * vi + 0];
        const float py = vertices[3 * vi + 1];
        const float pz = vertices[3 * vi + 2];

        const float rx = vx - px, ry = vy - py, rz = vz - pz;

        const float* __restrict__ r = R + 9 * ni;
        const float rotx = fmaf(r[0], rx, fmaf(r[1], ry, r[2] * rz));
        const float roty = fmaf(r[3], rx, fmaf(r[4], ry, r[5] * rz));
        const float rotz = fmaf(r[6], rx, fmaf(r[7], ry, r[8] * rz));

        const float tx = T[3 * ni + 0];
        const float ty = T[3 * ni + 1];
        const float tz = T[3 * ni + 2];

        const float w = weights[3 * n + k];
        ox = fmaf(w, rotx + px + tx, ox);
        oy = fmaf(w, roty + py + ty, oy);
        oz = fmaf(w, rotz + pz + tz, oz);
    }

    out[3 * n + 0] = ox;
    out[3 * n + 1] = oy;
    out[3 * n + 2] = oz;
}

// ---------------------------------------------------------------------------
// ARAP + smooth-rotation loss kernel: one thread per node (9 neighbors each).
// No early return: lanes past N_NODES contribute 0; WMMA runs with full EXEC.
// ---------------------------------------------------------------------------
__global__ void __launch_bounds__(256) arap_kernel(
    const float* __restrict__ vertices,
    const float* __restrict__ R,
    const float* __restrict__ T,
    const int*   __restrict__ nodes_idx,
    const int*   __restrict__ one_ring,
    double*      __restrict__ acc)
{
    const int m = blockIdx.x * blockDim.x + threadIdx.x;

    float arap = 0.f;
    float sr   = 0.f;

    if (m < N_NODES_C) {
        const int vi = nodes_idx[m];
        const float nix = vertices[3 * vi + 0];
        const float niy = vertices[3 * vi + 1];
        const float niz = vertices[3 * vi + 2];

        float rm[9];
#pragma unroll
        for (int q = 0; q < 9; ++q) rm[q] = R[9 * m + q];

        const float tmx = T[3 * m + 0];
        const float tmy = T[3 * m + 1];
        const float tmz = T[3 * m + 2];

        for (int e = 0; e < NEIGH_C; ++e) {
            const int j  = one_ring[m * NEIGH_C + e];
            const int vj = nodes_idx[j];

            const float njx = vertices[3 * vj + 0];
            const float njy = vertices[3 * vj + 1];
            const float njz = vertices[3 * vj + 2];

            const float relx = nix - njx;
            const float rely = niy - njy;
            const float relz = niz - njz;

            const float rotx = fmaf(rm[0], relx, fmaf(rm[1], rely, rm[2] * relz));
            const float roty = fmaf(rm[3], relx, fmaf(rm[4], rely, rm[5] * relz));
            const float rotz = fmaf(rm[6], relx, fmaf(rm[7], rely, rm[8] * relz));

            const float tjx = T[3 * j + 0];
            const float tjy = T[3 * j + 1];
            const float tjz = T[3 * j + 2];

            const float dx = nix + tmx - njx - tjx - rotx;
            const float dy = niy + tmy - njy - tjy - roty;
            const float dz = niz + tmz - njz - tjz - rotz;
            arap = fmaf(dx, dx, arap);
            arap = fmaf(dy, dy, arap);
            arap = fmaf(dz, dz, arap);

            const float* __restrict__ rj = R + 9 * j;
#pragma unroll
            for (int q = 0; q < 9; ++q) {
                const float s = rm[q] - rj[q];
                sr = fmaf(s, s, sr);
            }
        }
    }

    // Fully reconverged here (no lane returned) -> EXEC is all ones.
    wave_reduce2_wmma(arap, sr, acc + 0, acc + 1);
}

// ---------------------------------------------------------------------------
// Finalize: scale the two f64 sums and append them after the 3M warped floats.
// ---------------------------------------------------------------------------
__global__ void final_kernel(const double* __restrict__ acc, float* __restrict__ out) {
    if (threadIdx.x == 0 && blockIdx.x == 0) {
        out[3 * N_VERTS_C + 0] = (float)(acc[0] / (double)N_NODES_C);
        out[3 * N_VERTS_C + 1] = (float)(acc[1] / ((double)N_NODES_C * NEIGH_C * 9.0));
    }
}

extern "C" void kernel_launch(void* const* d_in, const int* in_sizes, int n_in,
                              void* d_out, int out_size, void* d_ws, size_t ws_size,
                              hipStream_t stream) {
    const float* vertices  = (const float*)d_in[0];  // (1M, 3)
    const float* R         = (const float*)d_in[1];  // (1, 500K, 3, 3)
    const float* T         = (const float*)d_in[2];  // (1, 500K, 3)
    const float* weights   = (const float*)d_in[3];  // (1M, 3)
    const int*   nodes_idx = (const int*)d_in[4];    // (500K,)
    const int*   infl_idx  = (const int*)d_in[5];    // (1M, 3)
    const int*   one_ring  = (const int*)d_in[6];    // (500K, 9)

    float*  out = (float*)d_out;                     // 3,000,002 floats
    double* acc = (double*)d_ws;                     // 2 doubles of scratch

    (void)in_sizes; (void)n_in; (void)out_size; (void)ws_size;

    init_kernel<<<1, 32, 0, stream>>>(acc);

    const int block = 256;
    const int gridWarp = (N_VERTS_C + block - 1) / block;   // 3907
    warp_kernel<<<gridWarp, block, 0, stream>>>(vertices, R, T, weights,
                                                nodes_idx, infl_idx, out);

    const int gridArap = (N_NODES_C + block - 1) / block;   // 1954
    arap_kernel<<<gridArap, block, 0, stream>>>(vertices, R, T,
                                                nodes_idx, one_ring, acc);

    final_kernel<<<1, 32, 0, stream>>>(acc, out);
}